// SpatialTransformer_43164421325500
// MI455X (gfx1250) — compile-verified
//
#include <hip/hip_runtime.h>

// 3-D spatial transformer (trilinear grid_sample, zeros padding,
// align_corners=True) for MI455X / gfx1250.
//
// src (2,1,160,192,224) f32, flow (2,3,160,192,224) f32.
// Bandwidth-bound: ~275 MB HBM traffic -> ~12 us at 23.3 TB/s, provided the
// VALU side stays under the roofline. Key choices:
//  * pix = grid + flow directly (the reference's normalize->denormalize is the
//    identity; trilinear interp is continuous in pix, so this is safe) --
//    removes 3 IEEE f32 divisions (~30 VALU ops) per voxel.
//  * 3-D launch grid: b,z,y-tile from blockIdx -> no integer divisions, and
//    uniform base pointers -> SGPR-base + 32-bit-offset gather addressing.
//  * zeros padding folded into the six 1-D weights via single unsigned
//    bound checks; clamps lower to v_med3_i32.
//  * flow/out streamed with non-temporal b128 ops so the 55 MB src volume
//    stays resident in the 192 MB L2 (gathers hit L2 after first touch).
//  * global_prefetch_b8 16 KB ahead on the three streaming flow planes.

typedef float v4f __attribute__((ext_vector_type(4)));

namespace {

constexpr int B     = 2;
constexpr int D     = 160;
constexpr int H     = 192;
constexpr int W     = 224;
constexpr int HW    = H * W;     // 43,008
constexpr int PLANE = D * HW;    // 6,881,280
constexpr int VEC   = 4;         // v4f along W (W % 4 == 0)
constexpr int TX    = W / VEC;   // 56 threads across a row
constexpr int TY    = 4;         // 4 rows per block -> 224 threads = 7 waves

}  // namespace

__global__ __launch_bounds__(TX* TY)
void warp3d_trilinear_kernel(const float* __restrict__ src,
                             const float* __restrict__ flow,
                             float* __restrict__ out) {
    const int x = threadIdx.x * VEC;                 // 0..220, step 4
    const int y = blockIdx.x * TY + threadIdx.y;     // 0..191
    const int z = blockIdx.y;                        // 0..159 (scalar)
    const int b = blockIdx.z;                        // 0..1   (scalar)

    const int r = z * HW + y * W + x;                // offset inside one volume
    const float* fb = flow + (size_t)b * 3 * PLANE;  // planes: z, y, x disp
    const float* sb = src + (size_t)b * PLANE;       // C == 1

    // Prefetch the streaming flow planes ~16 KB ahead (gfx1250
    // global_prefetch_b8; speculative, no counter traffic).
    constexpr int PF = 4096;  // elements
    if (r + PF < PLANE) {
        __builtin_prefetch(fb + r + PF,             0, 1);
        __builtin_prefetch(fb + PLANE + r + PF,     0, 1);
        __builtin_prefetch(fb + 2 * PLANE + r + PF, 0, 1);
    }

    // Non-temporal b128 loads: flow is read exactly once.
    v4f fz4 = __builtin_nontemporal_load((const v4f*)(fb + r));
    v4f fy4 = __builtin_nontemporal_load((const v4f*)(fb + PLANE + r));
    v4f fx4 = __builtin_nontemporal_load((const v4f*)(fb + 2 * PLANE + r));

    v4f o4;
#pragma unroll
    for (int j = 0; j < VEC; ++j) {
        // align_corners=True normalize->denormalize is the identity.
        float pz = (float)z + fz4[j];
        float py = (float)y + fy4[j];
        float px = (float)(x + j) + fx4[j];

        float zf = floorf(pz), yf = floorf(py), xf = floorf(px);
        int z0 = (int)zf, y0 = (int)yf, x0 = (int)xf;
        float fz = pz - zf, fy = py - yf, fx = px - xf;

        // 1-D weights with zeros padding folded in; one unsigned compare
        // covers both bounds of each check.
        float wz0 = ((unsigned)z0       < (unsigned)D) ? 1.0f - fz : 0.0f;
        float wz1 = ((unsigned)(z0 + 1) < (unsigned)D) ? fz        : 0.0f;
        float wy0 = ((unsigned)y0       < (unsigned)H) ? 1.0f - fy : 0.0f;
        float wy1 = ((unsigned)(y0 + 1) < (unsigned)H) ? fy        : 0.0f;
        float wx0 = ((unsigned)x0       < (unsigned)W) ? 1.0f - fx : 0.0f;
        float wx1 = ((unsigned)(x0 + 1) < (unsigned)W) ? fx        : 0.0f;

        // Clamped addresses (v_med3_i32); zero weight handles padding.
        int zc0 = min(max(z0, 0), D - 1), zc1 = min(max(z0 + 1, 0), D - 1);
        int yc0 = min(max(y0, 0), H - 1), yc1 = min(max(y0 + 1, 0), H - 1);
        int xc0 = min(max(x0, 0), W - 1), xc1 = min(max(x0 + 1, 0), W - 1);

        int zr0 = zc0 * HW, zr1 = zc1 * HW;
        int yr0 = yc0 * W,  yr1 = yc1 * W;
        int row00 = zr0 + yr0, row01 = zr0 + yr1;
        int row10 = zr1 + yr0, row11 = zr1 + yr1;

        float w00 = wz0 * wy0, w01 = wz0 * wy1;
        float w10 = wz1 * wy0, w11 = wz1 * wy1;

        float acc = (w00 * wx0) * sb[row00 + xc0];
        acc = fmaf(w00 * wx1, sb[row00 + xc1], acc);
        acc = fmaf(w01 * wx0, sb[row01 + xc0], acc);
        acc = fmaf(w01 * wx1, sb[row01 + xc1], acc);
        acc = fmaf(w10 * wx0, sb[row10 + xc0], acc);
        acc = fmaf(w10 * wx1, sb[row10 + xc1], acc);
        acc = fmaf(w11 * wx0, sb[row11 + xc0], acc);
        acc = fmaf(w11 * wx1, sb[row11 + xc1], acc);
        o4[j] = acc;
    }

    // Non-temporal b128 store: output is write-once, never re-read.
    __builtin_nontemporal_store(o4, (v4f*)(out + (size_t)b * PLANE + r));
}

extern "C" void kernel_launch(void* const* d_in, const int* in_sizes, int n_in,
                              void* d_out, int out_size, void* d_ws, size_t ws_size,
                              hipStream_t stream) {
    const float* src  = (const float*)d_in[0];
    const float* flow = (const float*)d_in[1];
    float*       out  = (float*)d_out;

    dim3 grid(H / TY, D, B);   // (48, 160, 2) = 15,360 blocks
    dim3 block(TX, TY);        // (56, 4) = 224 threads = 7 wave32 waves
    warp3d_trilinear_kernel<<<grid, block, 0, stream>>>(src, flow, out);
}